// QuantumConvLayer_88965952569856
// MI455X (gfx1250) — compile-verified
//
#include <hip/hip_runtime.h>
#include <math.h>

// MI455X / gfx1250 quantum statevector simulator.
// 16 qubits, 2 layers, batch 128. One workgroup (16 wave32) per batch element.
// State (2x fp32 x 65536) lives in d_ws (L2-resident, ping-pong buffers).
// Gates applied as 16x16 complex GEMMs on V_WMMA_F32_16X16X4_F32. Ring-CNOT
// layers are pure index permutations (prefix-XOR) fused into gathers /
// measurement signs. All passes ping-pong between buffers: no src/dst
// aliasing, so next-tile loads overlap current-tile WMMAs and stores.

#define NQ      16
#define NL      2
#define NSTATE  65536          // 2^16 amplitudes
#define TPB     512            // 16 waves of 32 -> 4 waves/SIMD on the WGP
#define NWAVES  (TPB / 32)
#define NTILES  256            // 4096 combos / 16 cols per tile
#define TILES_PER_WAVE (NTILES / NWAVES)   // 16

typedef float v2f __attribute__((ext_vector_type(2)));
typedef float v8f __attribute__((ext_vector_type(8)));

struct cpx { float re, im; };
__device__ __forceinline__ cpx cmul(cpx a, cpx b) {
  return { a.re * b.re - a.im * b.im, a.re * b.im + a.im * b.re };
}

// D = A(16x4) * B(4x16) + C   (fp32 WMMA, wave32)
__device__ __forceinline__ v8f wmma4(v2f a, v2f b, v8f c) {
  return __builtin_amdgcn_wmma_f32_16x16x4_f32(
      /*neg_a=*/false, a, /*neg_b=*/false, b,
      /*c_mod=*/(short)0, c, /*reuse_a=*/false, /*reuse_b=*/false);
}

// Ring CNOT layer CNOT(0,1)..CNOT(14,15),CNOT(15,0) as an index permutation.
// Sequential target-XOR => prefix-XOR of bits; wrap fixes bit 0 with parity.
__device__ __forceinline__ unsigned perm_fwd(unsigned b) {
  unsigned p = b ^ (b << 1);
  p ^= p << 2;  p ^= p << 4;  p ^= p << 8;
  p &= 0xFFFFu;
  return (p & 0xFFFEu) | ((b ^ (p >> 15)) & 1u);
}
__device__ __forceinline__ unsigned perm_inv(unsigned c) {
  unsigned u   = (c ^ (c << 1)) & 0xFFFFu;
  unsigned par = (c >> 15) & 1u;
  return (u & 0xFFFCu) | ((u ^ (par << 1)) & 2u) | ((c ^ par) & 1u);
}

// MODE: 0 = generate input from encoding tables (g==0, imag==0),
//       1 = normal pass, 2 = gather through perm_inv (fuses CNOT layer).
template <int MODE>
__device__ void apply_group(const float* __restrict__ sRe,
                            const float* __restrict__ sIm,
                            float* __restrict__ dRe,
                            float* __restrict__ dIm,
                            const float* Mr, const float* Mi,
                            const float (*T)[16], int g) {
  const int lane = threadIdx.x & 31;
  const int wave = threadIdx.x >> 5;
  const int half = lane >> 4;       // 0: lanes 0-15, 1: lanes 16-31
  const int col  = lane & 15;
  const int row  = lane & 15;       // A-matrix M index for this lane

  // A fragments: Mr, Mi, -Mi for the 4 K-chunks.
  // Layout (ISA 16x4 fp32 A): v0 = K{0|2}, v1 = K{1|3} by lane half.
  v2f ar[4], ai[4], an[4];
#pragma unroll
  for (int k = 0; k < 4; ++k) {
    int kk = 4 * k + 2 * half;
    ar[k][0] = Mr[row * 16 + kk];     ar[k][1] = Mr[row * 16 + kk + 1];
    ai[k][0] = Mi[row * 16 + kk];     ai[k][1] = Mi[row * 16 + kk + 1];
    an[k][0] = -ai[k][0];             an[k][1] = -ai[k][1];
  }

  const unsigned s   = (unsigned)g * 4u;
  const unsigned mlo = (1u << s) - 1u;

  for (int t0 = 0; t0 < TILES_PER_WAVE; ++t0) {
    const unsigned tt = (unsigned)(wave * TILES_PER_WAVE + t0);
    const unsigned cc = tt * 16u + (unsigned)col;          // combo id for this column
    const unsigned base2 = (cc & mlo) | ((cc >> s) << (s + 4)); // nibble g cleared

    // MODE 0 (g==0): per-lane, per-tile partial product T1*T2*T3 (constant over k).
    float t123 = 0.0f;
    if (MODE == 0) {
      t123 = T[1][(base2 >> 4) & 15] * T[2][(base2 >> 8) & 15] *
             T[3][(base2 >> 12) & 15];
    }

    v8f accR, accI;
#pragma unroll
    for (int i = 0; i < 8; ++i) { accR[i] = 0.0f; accI[i] = 0.0f; }

#pragma unroll
    for (int k = 0; k < 4; ++k) {
      const unsigned rowA = 4u * k + 2u * half;            // B rows for this lane
      unsigned iA = base2 + (rowA << s);
      unsigned iB = iA + (1u << s);
      v2f br, bi;
      if (MODE == 0) {
        br[0] = t123 * T[0][rowA];  br[1] = t123 * T[0][rowA + 1];
        bi[0] = 0.0f;               bi[1] = 0.0f;
      } else if (MODE == 2) {
        unsigned jA = perm_inv(iA), jB = perm_inv(iB);
        br[0] = sRe[jA];  br[1] = sRe[jB];
        bi[0] = sIm[jA];  bi[1] = sIm[jB];
      } else {
        br[0] = sRe[iA];  br[1] = sRe[iB];
        bi[0] = sIm[iA];  bi[1] = sIm[iB];
      }
      // Re += Mr*Xr - Mi*Xi ; Im += Mr*Xi + Mi*Xr
      accR = wmma4(ar[k], br, accR);
      accI = wmma4(ai[k], br, accI);
      if (MODE != 0) {
        accR = wmma4(an[k], bi, accR);
        accI = wmma4(ar[k], bi, accI);
      }
    }

    // D layout: vgpr v, lane -> row = v + 8*half, col = lane&15
#pragma unroll
    for (int v = 0; v < 8; ++v) {
      const unsigned r = (unsigned)v + 8u * half;
      const unsigned idx = base2 + (r << s);
      dRe[idx] = accR[v];
      dIm[idx] = accI[v];
    }
  }
}

__global__ __launch_bounds__(TPB) void qcirc_kernel(const float* __restrict__ xs,
                                                    const float* __restrict__ w,
                                                    float* __restrict__ out,
                                                    float* ws) {
  __shared__ float Mre[NL][4][16][16];   // fused 4-qubit group unitaries
  __shared__ float Mim[NL][4][16][16];
  __shared__ float Ure[NL][NQ][4];       // fused 1-qubit unitaries (row-major 2x2)
  __shared__ float Uim[NL][NQ][4];
  __shared__ float T[4][16];             // encoding product tables
  __shared__ float red[16][TPB + 1];     // measurement reduction (+1 pad vs bank conflicts)

  const int b   = blockIdx.x;
  const int tid = threadIdx.x;
  const float* x = xs + b * NQ;

  float* Are = ws + (size_t)b * (4u * NSTATE);
  float* Aim = Are + NSTATE;
  float* Bre = Aim + NSTATE;
  float* Bim = Bre + NSTATE;

  // ---- build fused per-qubit unitaries U = Rz(w2)*Ry(w1)*Rx(w0) ----
  if (tid < NL * NQ) {
    int l = tid >> 4, q = tid & 15;
    const float* wq = w + (l * NQ + q) * 3;
    float cx = cosf(wq[0] * 0.5f), sx = sinf(wq[0] * 0.5f);
    float cy = cosf(wq[1] * 0.5f), sy = sinf(wq[1] * 0.5f);
    float cz = cosf(wq[2] * 0.5f), sz = sinf(wq[2] * 0.5f);
    // Ry*Rx
    cpx m00 = {  cy * cx,  sy * sx };
    cpx m01 = { -sy * cx, -cy * sx };
    cpx m10 = {  sy * cx, -cy * sx };
    cpx m11 = {  cy * cx, -sy * sx };
    cpx e0 = { cz, -sz }, e1 = { cz, sz };   // Rz diagonal
    cpx u[4] = { cmul(e0, m00), cmul(e0, m01), cmul(e1, m10), cmul(e1, m11) };
#pragma unroll
    for (int i = 0; i < 4; ++i) { Ure[l][q][i] = u[i].re; Uim[l][q][i] = u[i].im; }
  }
  __syncthreads();

  // ---- kron 4 qubits -> 16x16 group matrices; encoding tables ----
  for (int e = tid; e < NL * 4 * 256; e += TPB) {
    int l = e >> 10, g = (e >> 8) & 3, r = (e >> 4) & 15, c = e & 15;
    cpx p = { 1.0f, 0.0f };
#pragma unroll
    for (int j = 0; j < 4; ++j) {
      int q = 4 * g + j;
      int idx = (((r >> j) & 1) << 1) | ((c >> j) & 1);
      cpx u = { Ure[l][q][idx], Uim[l][q][idx] };
      p = cmul(p, u);
    }
    Mre[l][g][r][c] = p.re;
    Mim[l][g][r][c] = p.im;
  }
  if (tid < 64) {
    int g = tid >> 4, n = tid & 15;
    float prod = 1.0f;
#pragma unroll
    for (int j = 0; j < 4; ++j) {
      float a = x[4 * g + j] * 1.57079632679489662f;   // x*pi, halved for RY(theta/2)
      prod *= ((n >> j) & 1) ? sinf(a) : cosf(a);
    }
    T[g][n] = prod;
  }
  __syncthreads();

  // ---- layer 0: encoding-generated apply (g=0), then g=1..3, all ping-pong ----
  apply_group<0>(Bre, Bim, Are, Aim, &Mre[0][0][0][0], &Mim[0][0][0][0], T, 0);
  __syncthreads();
  apply_group<1>(Are, Aim, Bre, Bim, &Mre[0][1][0][0], &Mim[0][1][0][0], T, 1);
  __syncthreads();
  apply_group<1>(Bre, Bim, Are, Aim, &Mre[0][2][0][0], &Mim[0][2][0][0], T, 2);
  __syncthreads();
  apply_group<1>(Are, Aim, Bre, Bim, &Mre[0][3][0][0], &Mim[0][3][0][0], T, 3);
  __syncthreads();
  // ---- layer 1: gather through CNOT permutation (g=0), then g=1..3 ----
  apply_group<2>(Bre, Bim, Are, Aim, &Mre[1][0][0][0], &Mim[1][0][0][0], T, 0);
  __syncthreads();
  apply_group<1>(Are, Aim, Bre, Bim, &Mre[1][1][0][0], &Mim[1][1][0][0], T, 1);
  __syncthreads();
  apply_group<1>(Bre, Bim, Are, Aim, &Mre[1][2][0][0], &Mim[1][2][0][0], T, 2);
  __syncthreads();
  apply_group<1>(Are, Aim, Bre, Bim, &Mre[1][3][0][0], &Mim[1][3][0][0], T, 3);
  __syncthreads();

  // ---- measurement: <Z_q> = sum_b |B[b]|^2 * sign_q(perm_fwd(b)) ----
  float acc[16];
#pragma unroll
  for (int q = 0; q < 16; ++q) acc[q] = 0.0f;

  for (int j = 0; j < NSTATE / TPB; ++j) {
    unsigned idx = (unsigned)(j * TPB + tid);
    float re = Bre[idx], im = Bim[idx];
    float p = re * re + im * im;
    unsigned c = perm_fwd(idx);
#pragma unroll
    for (int q = 0; q < 16; ++q) {
      int sgn = (int)((c >> q) & 1u) << 31;
      acc[q] += __int_as_float(__float_as_int(p) ^ sgn);
    }
  }
#pragma unroll
  for (int q = 0; q < 16; ++q) red[q][tid] = acc[q];
  __syncthreads();
  if (tid < 16) {
    float ssum = 0.0f;
    for (int t = 0; t < TPB; ++t) ssum += red[tid][t];   // fixed order: deterministic
    out[b * NQ + tid] = ssum;
  }
}

extern "C" void kernel_launch(void* const* d_in, const int* in_sizes, int n_in,
                              void* d_out, int out_size, void* d_ws, size_t ws_size,
                              hipStream_t stream) {
  const float* x = (const float*)d_in[0];   // (BATCH, 16) float32
  const float* w = (const float*)d_in[1];   // (2, 16, 3) float32
  float* out = (float*)d_out;               // (BATCH, 16) float32
  float* ws  = (float*)d_ws;                // 2 state buffers per batch elem (fp32 re/im)
  int batch = in_sizes[0] / NQ;
  qcirc_kernel<<<dim3(batch), dim3(TPB), 0, stream>>>(x, w, out, ws);
}